// LossMN_43061342110397
// MI455X (gfx1250) — compile-verified
//
#include <hip/hip_runtime.h>

// ---------------- problem constants (from reference) ----------------
#define S_        26
#define A_        5
#define M_        30
#define P_        (S_ * S_ * A_)     // 3380
#define D_        25                 // 5 + C(20)
#define CW_       16.0f              // 416 / 26
#define INV_CW_   (1.0f / 16.0f)
#define BATCH_    128
#define NTHREADS  256
#define NWAVES    (NTHREADS / 32)
#define IOU_TH_   0.6f

// batch split for occupancy: 4 blocks per batch, 512 blocks total
#define SPLIT_    4
#define CHUNK_    (P_ / SPLIT_)      // 845
#define NBLK_     (BATCH_ * SPLIT_)  // 512

// workspace layout (floats); total 31488 floats ~= 123 KB
#define WS_ARGV   0                   // [NBLK_*M_] per-chunk argmax value
#define WS_ARGI   (NBLK_ * M_)        // [NBLK_*M_] per-chunk argmax index (int)
#define WS_CONF   (2 * NBLK_ * M_)    // [NBLK_]    per-chunk conf partial
#define WS_BLOC   (WS_CONF + NBLK_)   // [BATCH_]   per-batch loc sum
#define WS_BCONF  (WS_BLOC + BATCH_)  // [BATCH_]   per-batch conf sum

typedef __attribute__((ext_vector_type(2))) float v2f;
typedef __attribute__((ext_vector_type(8))) float v8f;

// ---------------- f32 wave reduction via WMMA ----------------
// D = A(16x4) * ones(4x16): D[i][j] = rowsum_i. With A vgpr0 = v and vgpr1 = 0,
// rowsum_i = v_i + v_{i+16}; summing the 8 D vgprs leaves half-sums per 16-lane
// group; one xor-16 shuffle completes the 32-lane sum, all in f32.
__device__ __forceinline__ float wave_sum(float v) {
#if __has_builtin(__builtin_amdgcn_wmma_f32_16x16x4_f32)
    v2f a = {v, 0.0f};
    v2f b = {1.0f, 1.0f};
    v8f c = {};
    v8f d = __builtin_amdgcn_wmma_f32_16x16x4_f32(
        false, a, false, b, (short)0, c, false, false);
    float s = d[0] + d[1] + d[2] + d[3] + d[4] + d[5] + d[6] + d[7];
    s += __shfl_xor(s, 16, 32);
    return s;
#else
    #pragma unroll
    for (int off = 16; off > 0; off >>= 1) v += __shfl_xor(v, off, 32);
    return v;
#endif
}

__device__ __forceinline__ float fast_rcp(float x) {
#if __has_builtin(__builtin_amdgcn_rcpf)
    return __builtin_amdgcn_rcpf(x);
#else
    return 1.0f / x;
#endif
}

__device__ __forceinline__ float sigf(float x) {
    return 1.0f / (1.0f + __expf(-x));
}

// branchless (val,idx) argmax combine; ties keep the lower index (jnp.argmax)
__device__ __forceinline__ void amax_combine(float& v, int& i, float ov, int oi) {
    const bool take = (ov > v) | ((ov == v) & (oi < i));   // no short-circuit
    v = take ? ov : v;
    i = take ? oi : i;
}

// ---------------- stage 1: IoU scan over one chunk of one batch ----------------
__global__ void __launch_bounds__(NTHREADS)
yolo_loss_main(const float* __restrict__ pred,
               const float* __restrict__ targ,
               const float* __restrict__ anch,
               float* __restrict__ ws) {
    __shared__ float gx1[M_], gx2[M_], gy1[M_], gy2[M_], garea[M_];
    __shared__ int   gval[M_];
    __shared__ float anc[2 * A_];
    __shared__ float redv[NWAVES * M_];
    __shared__ int   redi[NWAVES * M_];
    __shared__ float redc[NWAVES];

    const int bid  = blockIdx.x;
    const int b    = bid / SPLIT_;
    const int q    = bid % SPLIT_;
    const int tid  = threadIdx.x;
    const int lane = tid & 31;
    const int wid  = tid >> 5;

    if (tid < 2 * A_) anc[tid] = anch[tid];
    if (tid < M_) {
        const float* t = targ + ((size_t)b * M_ + tid) * 5;
        float x1 = t[0], y1 = t[1], w = t[2], h = t[3], f = t[4];
        float xc = x1 + 0.5f * w;
        float yc = y1 + 0.5f * h;
        float gw = w - x1;          // faithful to source's gt width formula
        float gh = h * y1;          // faithful to source's gt height formula
        gx1[tid] = xc - 0.5f * gw;  gx2[tid] = xc + 0.5f * gw;
        gy1[tid] = yc - 0.5f * gh;  gy2[tid] = yc + 0.5f * gh;
        garea[tid] = gw * gh + 1e-9f;   // fold the +eps of the union here
        gval[tid]  = (f == 1.0f) ? 1 : 0;
    }
    __syncthreads();

    // per-m running argmax over this thread's preds (registers, unrolled)
    float bval[M_];
    int   bidx[M_];
    #pragma unroll
    for (int m = 0; m < M_; ++m) { bval[m] = -1.0f; bidx[m] = P_; }

    float confs = 0.0f;
    const int pstart = q * CHUNK_;
    const int pend   = pstart + CHUNK_;

    for (int p = pstart + tid; p < pend; p += NTHREADS) {
        const float* pp = pred + ((size_t)b * P_ + p) * D_;
        __builtin_prefetch(pp + (size_t)NTHREADS * D_, 0, 1);  // global_prefetch_b8
        float v0 = pp[0], v1 = pp[1], v2 = pp[2], v3 = pp[3], v4 = pp[4];

        int a   = p % A_;
        int col = (p / A_) % S_;
        int row = p / (A_ * S_);

        float plx = sigf(v0);
        float ply = sigf(v1);
        float plz = sigf(v2) * 0.5f;
        float plw = sigf(v3) * 0.5f;
        float pc  = sigf(v4);

        float bx = (plx + (float)col) * CW_;
        float by = (ply + (float)row) * CW_;
        float bw = __expf(plz) * anc[2 * a]     * CW_;
        float bh = __expf(plw) * anc[2 * a + 1] * CW_;
        float px1 = bx - 0.5f * bw, px2 = bx + 0.5f * bw;
        float py1 = by - 0.5f * bh, py2 = by + 0.5f * bh;
        float pa  = bw * bh;

        int obj = 0;
        #pragma unroll
        for (int m = 0; m < M_; ++m) {
            float dx = fminf(gx2[m], px2) - fmaxf(gx1[m], px1);
            float dy = fminf(gy2[m], py2) - fmaxf(gy1[m], py1);
            dx = fmaxf(dx, 0.0f);
            dy = fmaxf(dy, 0.0f);
            float inter = dx * dy;
            float uni   = garea[m] + pa - inter;   // includes the 1e-9 eps
            float iou   = inter * fast_rcp(uni);
            const bool gt = iou > bval[m];         // strict: first (lowest p) wins
            bval[m] = gt ? iou : bval[m];
            bidx[m] = gt ? p : bidx[m];
            obj |= (gval[m] & (int)(iou > IOU_TH_));
        }
        float dm = pc - 1.0f;
        confs += obj ? dm * dm : 0.5f * pc * pc;
    }

    // block-level argmax per annotation, branchless combines
    #pragma unroll
    for (int m = 0; m < M_; ++m) {
        float v = bval[m];
        int   i = bidx[m];
        #pragma unroll
        for (int off = 16; off > 0; off >>= 1) {
            float ov = __shfl_xor(v, off, 32);
            int   oi = __shfl_xor(i, off, 32);
            amax_combine(v, i, ov, oi);
        }
        if (lane == 0) { redv[wid * M_ + m] = v; redi[wid * M_ + m] = i; }
    }

    // block conf partial via WMMA wave reduce (uniform flow => EXEC all 1s)
    float csum = wave_sum(confs);
    if (lane == 0) redc[wid] = csum;
    __syncthreads();

    if (tid < M_) {
        float v = redv[tid];
        int   i = redi[tid];
        #pragma unroll
        for (int w = 1; w < NWAVES; ++w) {
            amax_combine(v, i, redv[w * M_ + tid], redi[w * M_ + tid]);
        }
        ws[WS_ARGV + bid * M_ + tid] = v;
        ((int*)(ws + WS_ARGI))[bid * M_ + tid] = i;
    }
    if (tid == 0) {
        float C = 0.0f;
        #pragma unroll
        for (int w = 0; w < NWAVES; ++w) C += redc[w];
        ws[WS_CONF + bid] = C;
    }
}

// ---------------- stage 2: one wave per batch -------------------------
// Combines chunk argmaxes, resolves duplicate responsible cells (highest m
// wins, emulating the scatter), computes the localization loss, and folds the
// chunk conf partials.
__global__ void __launch_bounds__(32)
yolo_loss_stage2(const float* __restrict__ pred,
                 const float* __restrict__ targ,
                 const float* __restrict__ anch,
                 float* __restrict__ ws) {
    const int b    = blockIdx.x;
    const int lane = threadIdx.x;
    const int* wsi = (const int*)(ws + WS_ARGI);

    // GT data for my annotation
    float xc = 0.0f, yc = 0.0f, wr = 1.0f, hr = 1.0f;
    int valid = 0;
    if (lane < M_) {
        const float* t = targ + ((size_t)b * M_ + lane) * 5;
        float x1 = t[0], y1 = t[1], w = t[2], h = t[3], f = t[4];
        xc = x1 + 0.5f * w;
        yc = y1 + 0.5f * h;
        wr = w; hr = h;
        valid = (f == 1.0f) ? 1 : 0;
    }

    // combine chunk argmaxes (chunks ordered by p => lowest-index tie-break)
    float v   = -1.0f;
    int   idx = P_;
    if (lane < M_) {
        v   = ws[WS_ARGV + (b * SPLIT_ + 0) * M_ + lane];
        idx = wsi[(b * SPLIT_ + 0) * M_ + lane];
        #pragma unroll
        for (int q = 1; q < SPLIT_; ++q) {
            amax_combine(v, idx,
                         ws[WS_ARGV + (b * SPLIT_ + q) * M_ + lane],
                         wsi[(b * SPLIT_ + q) * M_ + lane]);
        }
    }

    // duplicate resolution: among valid annotations mapping to the same slot,
    // the highest m owns it (deterministic emulation of the scatter)
    int owner = 1;
    #pragma unroll
    for (int j = 0; j < M_; ++j) {
        int oi = __shfl(idx,   j, 32);
        int ov = __shfl(valid, j, 32);
        const bool kill = (j > lane) & (ov != 0) & (oi == idx);
        owner = kill ? 0 : owner;
    }

    float locs = 0.0f;
    if (lane < M_ && valid && owner) {
        const float* pp = pred + ((size_t)b * P_ + idx) * D_;
        float plx = sigf(pp[0]);
        float ply = sigf(pp[1]);
        float plz = sigf(pp[2]) * 0.5f;
        float plw = sigf(pp[3]) * 0.5f;
        int ra = idx % A_;
        int rw = (idx / A_) % S_;
        int rh = idx / (A_ * S_);
        float tx = (xc - (float)rw * CW_) * INV_CW_;
        float ty = (yc - (float)rh * CW_) * INV_CW_;
        float tw = __logf((wr * INV_CW_) / anch[2 * ra]);
        float th = __logf((hr * INV_CW_) / anch[2 * ra + 1]);
        float e0 = plx - tx, e1 = ply - ty, e2 = plz - tw, e3 = plw - th;
        locs = e0 * e0 + e1 * e1 + e2 * e2 + e3 * e3;
    }

    float confp = (lane < SPLIT_) ? ws[WS_CONF + b * SPLIT_ + lane] : 0.0f;

    float L = wave_sum(locs);   // EXEC all 1s: 32-thread block, uniform flow
    float C = wave_sum(confp);
    if (lane == 0) {
        ws[WS_BLOC  + b] = L;
        ws[WS_BCONF + b] = C;
    }
}

// ---------------- finalize: reduce 128 batch partials to 3 scalars ----------------
__global__ void __launch_bounds__(BATCH_)
yolo_loss_final(const float* __restrict__ ws, float* __restrict__ out) {
    const int tid  = threadIdx.x;
    const int lane = tid & 31;
    const int wid  = tid >> 5;
    __shared__ float ra[BATCH_ / 32], rb[BATCH_ / 32];

    float L = ws[WS_BLOC  + tid];
    float C = ws[WS_BCONF + tid];
    float lw = wave_sum(L);
    float cw = wave_sum(C);
    if (lane == 0) { ra[wid] = lw; rb[wid] = cw; }
    __syncthreads();
    if (tid == 0) {
        float Ls = 0.0f, Cs = 0.0f;
        #pragma unroll
        for (int w = 0; w < BATCH_ / 32; ++w) { Ls += ra[w]; Cs += rb[w]; }
        Ls = 5.0f * Ls / (float)BATCH_;   // ALPHA_COORD / Bt
        Cs = Cs / (float)BATCH_;
        out[0] = Ls + Cs;
        out[1] = Ls;
        out[2] = Cs;
    }
}

// ---------------- entry ----------------
extern "C" void kernel_launch(void* const* d_in, const int* in_sizes, int n_in,
                              void* d_out, int out_size, void* d_ws, size_t ws_size,
                              hipStream_t stream) {
    const float* pred = (const float*)d_in[0];   // [128, 26, 26, 5, 25] f32
    const float* targ = (const float*)d_in[1];   // [128, 30, 5] f32
    const float* anch = (const float*)d_in[2];   // [5, 2] f32
    float* out = (float*)d_out;                  // [3] f32
    float* ws  = (float*)d_ws;                   // uses 31488 floats (~123 KB)

    yolo_loss_main  <<<NBLK_,  NTHREADS, 0, stream>>>(pred, targ, anch, ws);
    yolo_loss_stage2<<<BATCH_, 32,       0, stream>>>(pred, targ, anch, ws);
    yolo_loss_final <<<1,      BATCH_,   0, stream>>>(ws, out);
}